// GNNRegressor_54735063220225
// MI455X (gfx1250) — compile-verified
//
#include <hip/hip_runtime.h>
#include <hip/hip_bf16.h>

typedef __attribute__((ext_vector_type(2))) float v2f;
typedef __attribute__((ext_vector_type(8))) float v8f;

#define FEAT 128

// ---------------- degree / normalization ----------------

__global__ __launch_bounds__(256) void k_deg_init(float* deg, int n) {
    int i = blockIdx.x * 256 + threadIdx.x;
    if (i < n) deg[i] = 1.0f;               // self-loop contributes 1 to in-degree
}

__global__ __launch_bounds__(256) void k_deg_edges(const long long* __restrict__ dst,
                                                   float* deg, int ne) {
    int e = blockIdx.x * 256 + threadIdx.x;
    if (e < ne) atomicAdd(&deg[(int)dst[e]], 1.0f);
}

__global__ __launch_bounds__(256) void k_rsqrt(float* deg, int n) {
    int i = blockIdx.x * 256 + threadIdx.x;
    if (i < n) deg[i] = rsqrtf(deg[i]);     // deg >= 1 always (self-loop)
}

// ---------------- GEMM: Y[n,128] = X[n,128] @ W[128,128] via fp32 WMMA ----------------
// One block = 256 threads = 8 waves; block covers a 16-row strip, wave w covers cols [16w,16w+16).
// K marched in steps of 4 with V_WMMA_F32_16X16X4_F32 (exact fp32).

__global__ __launch_bounds__(256) void k_gemm128_wmma(const float* __restrict__ X,
                                                      const float* __restrict__ W,
                                                      float* __restrict__ Y, int nrows) {
    const int wave   = threadIdx.x >> 5;
    const int lane   = threadIdx.x & 31;
    const int laneLo = lane & 15;
    const int laneHi = lane >> 4;           // 0: K pair {k,k+1}, 1: {k+2,k+3}
    const int rowBase = blockIdx.x * 16;
    const int colBase = wave * 16;

    int arow = rowBase + laneLo;
    if (arow >= nrows) arow = nrows - 1;    // clamp for ragged last tile
    const float* __restrict__ aptr = X + (size_t)arow * FEAT;
    const int col = colBase + laneLo;

    v8f acc = {};
#pragma unroll 4
    for (int k = 0; k < FEAT; k += 4) {
        const int ka = k + 2 * laneHi;
        v2f a, b;
        a.x = aptr[ka];
        a.y = aptr[ka + 1];
        b.x = W[(size_t)ka * FEAT + col];
        b.y = W[(size_t)(ka + 1) * FEAT + col];
        // 8 args: (neg_a, A, neg_b, B, c_mod, C, reuse_a, reuse_b)
        acc = __builtin_amdgcn_wmma_f32_16x16x4_f32(false, a, false, b,
                                                    (short)0, acc, false, false);
    }

    // C/D layout: VGPR r holds M = r + 8*laneHi, N = laneLo
    const int rbase = rowBase + 8 * laneHi;
    float* __restrict__ yptr = Y + (size_t)rbase * FEAT + col;
    if (rowBase + 16 <= nrows) {
        // full tile: straight-line stores, row stride = FEAT floats
#pragma unroll
        for (int r = 0; r < 8; ++r)
            yptr[(size_t)r * FEAT] = acc[r];
    } else {
#pragma unroll
        for (int r = 0; r < 8; ++r)
            if (rbase + r < nrows) yptr[(size_t)r * FEAT] = acc[r];
    }
}

// ---------------- aggregation ----------------

// out[i,:] = hW[i,:] * dinv[i]^2   (self-loop term; also initializes the accumulator)
__global__ __launch_bounds__(256) void k_selfloop_init(const float* __restrict__ hW,
                                                       const float* __restrict__ dinv,
                                                       float* __restrict__ out, int n) {
    int t = blockIdx.x * 256 + threadIdx.x;          // one thread per 4 features
    int node = t >> 5;
    if (node >= n) return;
    int f4 = (t & 31);
    float di = dinv[node];
    float w = di * di;
    const float4* hp = (const float4*)(hW + (size_t)node * FEAT);
    float4 v = hp[f4];
    float4 o = make_float4(v.x * w, v.y * w, v.z * w, v.w * w);
    ((float4*)(out + (size_t)node * FEAT))[f4] = o;
}

// For each edge: out[dst,:] += hW[src,:] * dinv[src]*dinv[dst].  One wave per edge.
__global__ __launch_bounds__(256) void k_agg_edges(const long long* __restrict__ src,
                                                   const long long* __restrict__ dst,
                                                   const float* __restrict__ dinv,
                                                   const float* __restrict__ hW,
                                                   float* __restrict__ out, int ne) {
    int e = blockIdx.x * 8 + (threadIdx.x >> 5);
    if (e >= ne) return;
    int lane = threadIdx.x & 31;
    int s = (int)src[e];
    int d = (int)dst[e];
    float w = dinv[s] * dinv[d];
    const float4* hp = (const float4*)(hW + (size_t)s * FEAT);
    float4 v = hp[lane];
    float* op = out + (size_t)d * FEAT + lane * 4;
    atomicAdd(op + 0, v.x * w);
    atomicAdd(op + 1, v.y * w);
    atomicAdd(op + 2, v.z * w);
    atomicAdd(op + 3, v.w * w);
}

// h[i,f] = relu(acc[i,f] + b[f])
__global__ __launch_bounds__(256) void k_bias_relu(const float* __restrict__ acc,
                                                   const float* __restrict__ b,
                                                   float* __restrict__ h, int n) {
    int t = blockIdx.x * 256 + threadIdx.x;
    int node = t >> 5;
    if (node >= n) return;
    int f4 = (t & 31);
    float4 v = ((const float4*)(acc + (size_t)node * FEAT))[f4];
    float4 bb = ((const float4*)b)[f4];
    float4 o;
    o.x = fmaxf(v.x + bb.x, 0.0f);
    o.y = fmaxf(v.y + bb.y, 0.0f);
    o.z = fmaxf(v.z + bb.z, 0.0f);
    o.w = fmaxf(v.w + bb.w, 0.0f);
    ((float4*)(h + (size_t)node * FEAT))[f4] = o;
}

// out[n] = dot(h[n,:], Wl) + bl   — one wave per node, shuffle reduction
__global__ __launch_bounds__(256) void k_head(const float* __restrict__ h,
                                              const float* __restrict__ Wl,
                                              const float* __restrict__ bl,
                                              float* __restrict__ out, int n) {
    int node = blockIdx.x * 8 + (threadIdx.x >> 5);
    if (node >= n) return;
    int lane = threadIdx.x & 31;
    float4 a = ((const float4*)(h + (size_t)node * FEAT))[lane];
    float4 w = ((const float4*)Wl)[lane];
    float sum = a.x * w.x + a.y * w.y + a.z * w.z + a.w * w.w;
#pragma unroll
    for (int off = 16; off > 0; off >>= 1)
        sum += __shfl_xor(sum, off, 32);
    if (lane == 0) out[node] = sum + bl[0];
}

// ---------------- launch ----------------

extern "C" void kernel_launch(void* const* d_in, const int* in_sizes, int n_in,
                              void* d_out, int out_size, void* d_ws, size_t ws_size,
                              hipStream_t stream) {
    const float*     x   = (const float*)d_in[0];
    const long long* ei  = (const long long*)d_in[1];   // int64 [2, E]
    const float*     W1  = (const float*)d_in[2];
    const float*     b1  = (const float*)d_in[3];
    const float*     W2  = (const float*)d_in[4];
    const float*     b2  = (const float*)d_in[5];
    const float*     Wl  = (const float*)d_in[6];
    const float*     bl  = (const float*)d_in[7];
    float*           out = (float*)d_out;

    const int N = in_sizes[0] / FEAT;       // 100000
    const int E = in_sizes[1] / 2;          // 1600000
    const long long* src = ei;
    const long long* dst = ei + E;

    // workspace layout: dinv | bufA (N*128 f32) | bufB (N*128 f32)
    char* ws = (char*)d_ws;
    const size_t featBytes = (size_t)N * FEAT * sizeof(float);
    float* dinv = (float*)ws;
    float* bufA = (float*)(ws + ((((size_t)N * sizeof(float)) + 255) & ~(size_t)255));
    float* bufB = (float*)((char*)bufA + featBytes);

    const int nodeBlocks  = (N + 255) / 256;
    const int edgeBlocks  = (E + 255) / 256;
    const int gemmBlocks  = (N + 15) / 16;
    const int featBlocks  = ((N * 32) + 255) / 256;   // 1 thread per 4 features
    const int waveNodeBlk = (N + 7) / 8;              // 1 wave per node
    const int waveEdgeBlk = (E + 7) / 8;              // 1 wave per edge

    // degree + normalization
    k_deg_init <<<nodeBlocks, 256, 0, stream>>>(dinv, N);
    k_deg_edges<<<edgeBlocks, 256, 0, stream>>>(dst, dinv, E);
    k_rsqrt    <<<nodeBlocks, 256, 0, stream>>>(dinv, N);

    // layer 1: hW1 -> bufA, agg -> bufB, h1 -> bufA
    k_gemm128_wmma <<<gemmBlocks, 256, 0, stream>>>(x, W1, bufA, N);
    k_selfloop_init<<<featBlocks, 256, 0, stream>>>(bufA, dinv, bufB, N);
    k_agg_edges    <<<waveEdgeBlk, 256, 0, stream>>>(src, dst, dinv, bufA, bufB, E);
    k_bias_relu    <<<featBlocks, 256, 0, stream>>>(bufB, b1, bufA, N);

    // layer 2: hW2 -> bufB, agg -> bufA, h2 -> bufB
    k_gemm128_wmma <<<gemmBlocks, 256, 0, stream>>>(bufA, W2, bufB, N);
    k_selfloop_init<<<featBlocks, 256, 0, stream>>>(bufB, dinv, bufA, N);
    k_agg_edges    <<<waveEdgeBlk, 256, 0, stream>>>(src, dst, dinv, bufB, bufA, E);
    k_bias_relu    <<<featBlocks, 256, 0, stream>>>(bufA, b2, bufB, N);

    // head
    k_head<<<waveNodeBlk, 256, 0, stream>>>(bufB, Wl, bl, out, N);
}